// SpectralConv1d_wave_24592982737256
// MI455X (gfx1250) — compile-verified
//
#include <hip/hip_runtime.h>
#include <math.h>

typedef __attribute__((ext_vector_type(2))) float v2f;
typedef __attribute__((ext_vector_type(8))) float v8f;

#define T_SEQ     10
#define D_MODEL   256
#define BATCH     16
#define RES       200
#define MODES_OUT 16
#define MODES_IN  32
#define XPAD      208   // 13 x-tiles of 16 (200 padded)
#define NXT       13
#define KDIM      64    // 2 * MODES_IN (real half | imag half), contiguous in v

// ---------------------------------------------------------------------------
// Kernel 1: fused spectral weight  Wf[t][x][k], k in [0,64)
//   Wf[t,x,m]    =  sum_o ( wr[t,o,m]*cos(2*pi*o*x/200) - wi[t,o,m]*sin(...) )
//   Wf[t,x,32+m] = -sum_o ( wi[t,o,m]*cos(...)          + wr[t,o,m]*sin(...) )
// Pad rows (x >= 200) are zeroed so the GEMM can run un-predicated.
// ---------------------------------------------------------------------------
__global__ __launch_bounds__(256) void build_wf(const float* __restrict__ w_real,
                                                const float* __restrict__ w_imag,
                                                float* __restrict__ wf) {
  int idx = blockIdx.x * blockDim.x + threadIdx.x;   // T_SEQ*XPAD*32 threads
  int m = idx & 31;
  int x = (idx >> 5) % XPAD;
  int t = idx / (XPAD * 32);
  if (t >= T_SEQ) return;

  float accr = 0.0f, acci = 0.0f;
  if (x < RES) {
    const float base = (6.283185307179586476925f / 200.0f) * (float)x;
    #pragma unroll
    for (int o = 0; o < MODES_OUT; ++o) {
      float s, c;
      sincosf(base * (float)o, &s, &c);
      float wr = w_real[(t * MODES_OUT + o) * MODES_IN + m];
      float wi = w_imag[(t * MODES_OUT + o) * MODES_IN + m];
      accr += wr * c - wi * s;
      acci -= wi * c + wr * s;
    }
  }
  float* row = wf + (t * XPAD + x) * KDIM;
  row[m]            = accr;
  row[MODES_IN + m] = acci;
}

// ---------------------------------------------------------------------------
// Kernel 2: out[b,t,x,h] = sum_k V2[b,t,h,k] * Wf[t,x,k]   via f32 WMMA.
// One wave computes one 16(x) x 32(h) output tile: two chained accumulators
// over K=64 (16 steps of v_wmma_f32_16x16x4_f32), A-fragment reused.
//
// f32 fragment layouts (lane l, lm=l&15, lh=l>>4, K-map = vgpr + 2*lh):
//   A 16x4 : {Wf[x0+lm][4kb+2lh], Wf[x0+lm][4kb+2lh+1]}  -> contiguous float2
//   B 4x16 : {V2[h0+lm][4kb+2lh], V2[h0+lm][4kb+2lh+1]}  -> contiguous float2
//   D 16x16: vgpr j -> out[x0 + j + 8*lh][h0 + lm]
// ---------------------------------------------------------------------------
__global__ __launch_bounds__(256) void spectral_wmma(const float* __restrict__ v,
                                                     const float* __restrict__ wf,
                                                     float* __restrict__ out) {
  const int lane = threadIdx.x & 31;
  const int wid  = (int)((blockIdx.x * blockDim.x + threadIdx.x) >> 5);

  const int hp = wid & 7;                 // 8 pairs of h-tiles (32 h each)
  const int xt = (wid >> 3) % NXT;        // 13 x-tiles
  const int tt = (wid / (8 * NXT)) % T_SEQ;
  const int b  = wid / (8 * NXT * T_SEQ);
  if (b >= BATCH) return;

  const int lm = lane & 15;
  const int lh = lane >> 4;

  const float* arow  = wf + (size_t)(tt * XPAD + xt * 16 + lm) * KDIM + 2 * lh;
  const float* brow0 = v  + (size_t)(((b * T_SEQ + tt) * D_MODEL) + hp * 32 + lm) * KDIM + 2 * lh;
  const float* brow1 = brow0 + 16 * KDIM;

  v8f acc0 = {};
  v8f acc1 = {};
  #pragma unroll
  for (int kb = 0; kb < 16; ++kb) {
    v2f a  = *(const v2f*)(arow  + 4 * kb);
    v2f b0 = *(const v2f*)(brow0 + 4 * kb);
    v2f b1 = *(const v2f*)(brow1 + 4 * kb);
    acc0 = __builtin_amdgcn_wmma_f32_16x16x4_f32(false, a, false, b0,
                                                 (short)0, acc0, false, false);
    acc1 = __builtin_amdgcn_wmma_f32_16x16x4_f32(false, a, false, b1,
                                                 (short)0, acc1, false, false);
  }

  float* obase = out + (size_t)(b * T_SEQ + tt) * RES * D_MODEL + hp * 32 + lm;
  const int xb = xt * 16;
  #pragma unroll
  for (int j = 0; j < 8; ++j) {
    int x = xb + j + 8 * lh;
    if (x < RES) {                         // diverges only on the last x-tile
      obase[(size_t)x * D_MODEL]      = acc0[j];
      obase[(size_t)x * D_MODEL + 16] = acc1[j];
    }
  }
}

// ---------------------------------------------------------------------------
extern "C" void kernel_launch(void* const* d_in, const int* in_sizes, int n_in,
                              void* d_out, int out_size, void* d_ws, size_t ws_size,
                              hipStream_t stream) {
  const float* v      = (const float*)d_in[0];   // (16,10,256,2,32) f32
  const float* w_real = (const float*)d_in[1];   // (10,16,32) f32
  const float* w_imag = (const float*)d_in[2];   // (10,16,32) f32
  float* out = (float*)d_out;                    // (16,10,200,256) f32
  float* wf  = (float*)d_ws;                     // needs 10*208*64*4 = 520 KB

  // Kernel 1: T_SEQ*XPAD*32 = 66,560 threads
  build_wf<<<(T_SEQ * XPAD * 32) / 256, 256, 0, stream>>>(w_real, w_imag, wf);

  // Kernel 2: BATCH*T_SEQ*13*8 = 16,640 waves, 8 waves/block
  spectral_wmma<<<(BATCH * T_SEQ * NXT * 8) / 8, 256, 0, stream>>>(v, wf, out);
}